// PPNProp_3178275799596
// MI455X (gfx1250) — compile-verified
//
#include <hip/hip_runtime.h>
#include <hip/hip_bf16.h>

// PPNProp: dropout on sparse adjacency values + SpMM (out[dst] += w_e * x[src]).
// Atomic/L2-bandwidth bound on MI455X: x (25.6MB) is L2-resident (192MB L2);
// edge lists stream through CDNA5 async global->LDS DMA (ASYNCcnt, double
// buffered); scatter uses native global_atomic_add_f32 (forced via asm).

#define WAVE   32
#define BLOCK  256
#define CHUNK  256   // edges staged per block iteration
#define FEAT   64    // D

// --- deterministic inverted-dropout mask (keep = 0.5, scale = 2.0) ----------
__device__ __forceinline__ unsigned edge_hash(unsigned v) {
    v += 42u * 0x9E3779B9u;         // "key 42"
    v ^= v >> 16; v *= 0x7feb352dU;
    v ^= v >> 15; v *= 0x846ca68bU;
    v ^= v >> 16;
    return v;
}

// Low 32 bits of a generic pointer to __shared__ are the wave-relative LDS
// byte address.
__device__ __forceinline__ unsigned lds_u32addr(const void* p) {
    return (unsigned)(unsigned long long)(p);
}

// CDNA5 async DMA: GLOBAL_LOAD_ASYNC_TO_LDS_B32, GVS mode (saddr + v_off).
__device__ __forceinline__ void async_ld_b32(unsigned lds, unsigned boff, const void* base) {
    asm volatile("global_load_async_to_lds_b32 %0, %1, %2 offset:0"
                 :
                 : "v"(lds), "v"(boff), "s"(base)
                 : "memory");
}

__device__ __forceinline__ void wait_async0() {
    asm volatile("s_wait_asynccnt 0" ::: "memory");
}

// Native f32 atomic add, no return (STOREcnt), GVS mode: base SGPR pair +
// signed 32-bit VGPR byte offset. Guarantees the L2 atomic unit is used.
__device__ __forceinline__ void atomic_add_f32(unsigned boff, float v, void* base) {
    asm volatile("global_atomic_add_f32 %0, %1, %2 offset:0"
                 :
                 : "v"(boff), "v"(v), "s"(base)
                 : "memory");
}

__global__ __launch_bounds__(BLOCK) void ppnp_spmm_kernel(
    const float* __restrict__ x,        // [N, 64]
    const float* __restrict__ w,        // [E]
    const int*   __restrict__ dst_idx,  // edge_index[0] : output row
    const int*   __restrict__ src_idx,  // edge_index[1] : gathered row
    float*       __restrict__ out,      // [N, 64]
    int E)
{
    __shared__ int   sSrc[2][CHUNK];
    __shared__ int   sDst[2][CHUNK];
    __shared__ float sW[2][CHUNK];

    const int tid  = threadIdx.x;
    const int lane = tid & (WAVE - 1);
    const int wid  = tid >> 5;                      // 0..7 (wave32)
    const int numChunks = (E + CHUNK - 1) / CHUNK;

    int c = blockIdx.x;
    if (c >= numChunks) return;

    // Prefetch first chunk into buffer 0 (clamp tail loads in-bounds).
    {
        int e = c * CHUNK + tid;
        unsigned off = (unsigned)(min(e, E - 1)) * 4u;
        async_ld_b32(lds_u32addr(&sSrc[0][tid]), off, src_idx);
        async_ld_b32(lds_u32addr(&sDst[0][tid]), off, dst_idx);
        async_ld_b32(lds_u32addr(&sW[0][tid]),   off, w);
    }

    int buf = 0;
    for (; c < numChunks; c += gridDim.x) {
        // Our wave's DMA into sXXX[buf] done; barrier publishes every wave's
        // DMA and fences last iteration's compute on the other buffer.
        wait_async0();
        __syncthreads();

        // Kick off DMA for the next chunk into the other buffer.
        const int cn = c + gridDim.x;
        if (cn < numChunks) {
            int e = cn * CHUNK + tid;
            unsigned off = (unsigned)(min(e, E - 1)) * 4u;
            const int nb = buf ^ 1;
            async_ld_b32(lds_u32addr(&sSrc[nb][tid]), off, src_idx);
            async_ld_b32(lds_u32addr(&sDst[nb][tid]), off, dst_idx);
            async_ld_b32(lds_u32addr(&sW[nb][tid]),   off, w);
        }

        // Wave `wid` owns 32 edges; all 32 lanes cooperate on one edge
        // (lane -> float2 slice of the 64-float row; 256B coalesced gather).
        const int chunkStart = c * CHUNK;
        const int cnt  = min(CHUNK, E - chunkStart);
        const int eBeg = wid * WAVE;
        const int eEnd = min(eBeg + WAVE, cnt);
        for (int e = eBeg; e < eEnd; ++e) {
            const int   src = sSrc[buf][e];
            const int   dst = sDst[buf][e];
            const float wv  = sW[buf][e];

            const unsigned h = edge_hash((unsigned)(chunkStart + e));
            const float scale = (h & 0x80000000u) ? 0.0f : 2.0f;  // keep p=0.5
            const float wk = wv * scale;

            const float2 v = ((const float2*)(x + (size_t)src * FEAT))[lane];
            const unsigned obase = (unsigned)dst * (FEAT * 4u) + (unsigned)lane * 8u;
            atomic_add_f32(obase + 0u, v.x * wk, out);
            atomic_add_f32(obase + 4u, v.y * wk, out);
        }
        buf ^= 1;
    }
}

__global__ void ppnp_zero_kernel(float* __restrict__ out, int n) {
    const int stride = 1024 * 256;
    for (int i = blockIdx.x * 256 + threadIdx.x; i < n; i += stride) out[i] = 0.0f;
}

extern "C" void kernel_launch(void* const* d_in, const int* in_sizes, int n_in,
                              void* d_out, int out_size, void* d_ws, size_t ws_size,
                              hipStream_t stream) {
    (void)n_in; (void)d_ws; (void)ws_size;
    const float* x  = (const float*)d_in[0];     // [N,64] f32
    const float* w  = (const float*)d_in[1];     // [E]    f32
    const int*   ei = (const int*)d_in[2];       // [2,E]  int (row0=dst, row1=src)
    const int E = in_sizes[1];
    float* out = (float*)d_out;

    // d_out is poisoned by the harness: zero it first (stream-ordered).
    ppnp_zero_kernel<<<1024, 256, 0, stream>>>(out, out_size);

    const int numChunks = (E + CHUNK - 1) / CHUNK;
    const int blocks = numChunks < 2048 ? numChunks : 2048;
    ppnp_spmm_kernel<<<blocks, BLOCK, 0, stream>>>(x, w, ei, ei + E, out, E);
}